// GMABSE3_20031727468995
// MI455X (gfx1250) — compile-verified
//
#include <hip/hip_runtime.h>
#include <math.h>

#define N_NODES 10000
#define N_EDGES 320000
#define NH 8

// ---------- order-preserving float <-> unsigned key (for atomic max) ----------
__device__ __forceinline__ unsigned fkey(float x) {
  unsigned u = __float_as_uint(x);
  return (u & 0x80000000u) ? ~u : (u | 0x80000000u);
}
__device__ __forceinline__ float fdecode(unsigned k) {
  return __uint_as_float((k & 0x80000000u) ? (k ^ 0x80000000u) : ~k);
}

typedef float v2f __attribute__((ext_vector_type(2)));
typedef float v8f __attribute__((ext_vector_type(8)));

// ---------- CDNA5 async global->LDS path (guarded; fallback = b128 copy) ----------
#if defined(__has_builtin)
#if __has_builtin(__builtin_amdgcn_global_load_async_to_lds_b128) && \
    __has_builtin(__builtin_amdgcn_s_wait_asynccnt)
#define USE_ASYNC_LDS 1
#endif
#endif

#if USE_ASYNC_LDS
// builtin signature: (v4i __device__(AS1)*, v4i __shared__(AS3)*, imm offset, imm cpol)
typedef int vi4 __attribute__((__vector_size__(16)));
typedef __attribute__((address_space(1))) vi4 gvi4;
typedef __attribute__((address_space(3))) vi4 lvi4;
#endif

__device__ __forceinline__ void copy16_to_lds(const void* g, void* l) {
#if USE_ASYNC_LDS
  // AS3 pointers are 32-bit LDS offsets == low 32 bits of a generic LDS pointer.
  __builtin_amdgcn_global_load_async_to_lds_b128(
      (gvi4*)(unsigned long long)g,
      (lvi4*)(unsigned)(unsigned long long)l, 0, 0);
#else
  *(float4*)l = *(const float4*)g;
#endif
}

// ---------- init: zero outputs/denominators, set max-keys to key(-inf) ----------
__global__ void se3_init(float* __restrict__ out, float* __restrict__ denom,
                         unsigned* __restrict__ mkey) {
  int i = blockIdx.x * 256 + threadIdx.x;
  if (i < N_NODES * 256) out[i] = 0.0f;
  if (i < N_NODES * NH) { denom[i] = 0.0f; mkey[i] = 0x007FFFFFu; /* key(-inf) */ }
}

// ---------- pass 1: attention logits via WMMA segmented reduction ----------
// One wave32 handles 16 edges. Elementwise products p = kh * qh[dst] are staged
// in LDS (row pitch 260 -> conflict-free b64 reads), then reduced per head with
// V_WMMA_F32_16X16X4_F32 using a one-hot selector B (head constant per K-step of 4).
__global__ void __launch_bounds__(128)
se3_scores(const float* __restrict__ k0, const float* __restrict__ k1,
           const float* __restrict__ q0, const float* __restrict__ q1,
           const int* __restrict__ dst,
           float* __restrict__ ews, unsigned* __restrict__ mkey) {
  __shared__ float p[4][16][260];  // 4 waves x 16 edges x 256(+4 pad) products
  const int tid = threadIdx.x;
  const int lane = tid & 31;
  const int w = tid >> 5;
  const int eb = (blockIdx.x * 4 + w) * 16;

  // Stage elementwise products. Per iteration the whole wave covers half an
  // edge row (32 float4 lanes), so dst[e] is wave-uniform and k/q reads coalesce.
  for (int it = 0; it < 32; ++it) {
    int g = it * 32 + lane;          // float4 id within the 16x64 tile
    int r = g >> 6;                  // edge row 0..15
    int d = (g & 63) << 2;           // feature index 0..255, step 4
    int e = eb + r;
    int n = dst[e];
    int h = d >> 5;                  // head = d/32
    int j = d & 31;                  // within-head feature
    const float4* kp;
    const float4* qp;
    if (j < 8) {                     // degree-0 fiber (8 floats/head)
      kp = (const float4*)(k0 + (size_t)e * 64 + 8 * h + j);
      qp = (const float4*)(q0 + (size_t)n * 64 + 8 * h + j);
    } else {                         // degree-1 fiber (24 floats/head)
      kp = (const float4*)(k1 + (size_t)e * 192 + 24 * h + (j - 8));
      qp = (const float4*)(q1 + (size_t)n * 192 + 24 * h + (j - 8));
    }
    float4 kv = *kp, qv = *qp;
    float4 pv;
    pv.x = kv.x * qv.x; pv.y = kv.y * qv.y; pv.z = kv.z * qv.z; pv.w = kv.w * qv.w;
    *(float4*)&p[w][r][d] = pv;
  }
  __syncthreads();

  // WMMA reduction: A[16x4] = product chunk, B[4x16] = one-hot head selector.
  // A layout: lane<16 holds K=d,d+1 ; lane>=16 holds K=d+2,d+3 for row M=lane%16.
  const int r = lane & 15;
  const int hf = lane >> 4;
  const float* prow = &p[w][r][0];
  v8f acc0 = {0, 0, 0, 0, 0, 0, 0, 0};
  v8f acc1 = {0, 0, 0, 0, 0, 0, 0, 0};
#pragma unroll
  for (int h = 0; h < NH; ++h) {
    v2f b;
    b[0] = b[1] = (r == h) ? 1.0f : 0.0f;   // column lane%16 == head h
#pragma unroll
    for (int s = 0; s < 8; s += 2) {        // 8 K-steps of 4 per head
      v2f a0 = *(const v2f*)(prow + (h * 8 + s) * 4 + 2 * hf);
      v2f a1 = *(const v2f*)(prow + (h * 8 + s + 1) * 4 + 2 * hf);
      acc0 = __builtin_amdgcn_wmma_f32_16x16x4_f32(false, a0, false, b,
                                                   (short)0, acc0, false, false);
      acc1 = __builtin_amdgcn_wmma_f32_16x16x4_f32(false, a1, false, b,
                                                   (short)0, acc1, false, false);
    }
  }

  // C layout: lane holds column N=lane%16, rows M = v + 8*(lane>=16), v=0..7.
  const int col = r;
  const int mb = hf * 8;
  if (col < NH) {
#pragma unroll
    for (int v = 0; v < 8; ++v) {
      int e = eb + mb + v;
      float ev = (acc0[v] + acc1[v]) * 0.0625f;  // 1/sqrt(256)
      ews[(size_t)e * NH + col] = ev;
      int n = dst[e];
      atomicMax(&mkey[n * NH + col], fkey(ev));
    }
  }
}

// ---------- pass 2: exp + unnormalized weighted scatter ----------
// Block = 32 edges x 8 heads. v-tile staged in LDS via async-to-LDS DMA.
__global__ void __launch_bounds__(256)
se3_agg(const float* __restrict__ ews, const int* __restrict__ dst,
        const unsigned* __restrict__ mkey, float* __restrict__ denom,
        const float* __restrict__ v0, const float* __restrict__ v1,
        float* __restrict__ out0, float* __restrict__ out1) {
  __shared__ float sv0[32 * 64];    // 8 KB
  __shared__ float sv1[32 * 192];   // 24 KB
  const int tid = threadIdx.x;
  const int eb = blockIdx.x * 32;

  const float* gv0 = v0 + (size_t)eb * 64;
  const float* gv1 = v1 + (size_t)eb * 192;
#pragma unroll
  for (int c = 0; c < 8; ++c) {
    int i = c * 256 + tid;          // 2048 float4s total (512 v0 + 1536 v1)
    if (i < 512) copy16_to_lds((const float4*)gv0 + i, (float4*)sv0 + i);
    else         copy16_to_lds((const float4*)gv1 + (i - 512), (float4*)sv1 + (i - 512));
  }
#if USE_ASYNC_LDS
  __builtin_amdgcn_s_wait_asynccnt(0);
#endif
  __syncthreads();

  const int el = tid >> 3;          // local edge 0..31
  const int h = tid & 7;            // head
  const int e = eb + el;
  const int n = dst[e];
  const float m = fdecode(mkey[n * NH + h]);
  const float ex = expf(ews[(size_t)e * NH + h] - m);
  atomicAdd(&denom[n * NH + h], ex);

  const float* s0 = sv0 + el * 64 + 8 * h;
  float* o0 = out0 + (size_t)n * 64 + 8 * h;
#pragma unroll
  for (int i = 0; i < 8; ++i) atomicAdd(&o0[i], ex * s0[i]);

  const float* s1 = sv1 + el * 192 + 24 * h;
  float* o1 = out1 + (size_t)n * 192 + 24 * h;
#pragma unroll
  for (int i = 0; i < 24; ++i) atomicAdd(&o1[i], ex * s1[i]);
}

// ---------- pass 3: normalize by softmax denominator (per node/head) ----------
__global__ void se3_norm(float* __restrict__ out, const float* __restrict__ denom) {
  int idx = blockIdx.x * 256 + threadIdx.x;
  if (idx >= N_NODES * 256) return;
  int n, h;
  if (idx < N_NODES * 64) {         // out0: [N,64,1], head = (i%64)/8
    n = idx >> 6;
    h = (idx & 63) >> 3;
  } else {                          // out1: [N,64,3] flat, head = (j%192)/24
    int j = idx - N_NODES * 64;
    n = j / 192;
    h = (j % 192) / 24;
  }
  float dsum = denom[n * NH + h];
  out[idx] = (dsum > 0.0f) ? out[idx] / dsum : 0.0f;  // empty segments -> 0
}

extern "C" void kernel_launch(void* const* d_in, const int* in_sizes, int n_in,
                              void* d_out, int out_size, void* d_ws, size_t ws_size,
                              hipStream_t stream) {
  const float* v0 = (const float*)d_in[0];
  const float* v1 = (const float*)d_in[1];
  const float* k0 = (const float*)d_in[2];
  const float* k1 = (const float*)d_in[3];
  const float* q0 = (const float*)d_in[4];
  const float* q1 = (const float*)d_in[5];
  const int* dst = (const int*)d_in[6];

  // workspace layout: e logits [E*8] | max keys [N*8] | denom [N*8]  (~10.9 MB)
  float* ews = (float*)d_ws;
  unsigned* mkey = (unsigned*)(ews + (size_t)N_EDGES * NH);
  float* denom = (float*)(mkey + (size_t)N_NODES * NH);

  float* out = (float*)d_out;
  float* out0 = out;                              // [N,64]
  float* out1 = out + (size_t)N_NODES * 64;       // [N,64,3]

  se3_init<<<(N_NODES * 256 + 255) / 256, 256, 0, stream>>>(out, denom, mkey);
  se3_scores<<<N_EDGES / 64, 128, 0, stream>>>(k0, k1, q0, q1, dst, ews, mkey);
  se3_agg<<<N_EDGES / 32, 256, 0, stream>>>(ews, dst, mkey, denom, v0, v1, out0, out1);
  se3_norm<<<(N_NODES * 256 + 255) / 256, 256, 0, stream>>>(out, denom);
}